// WaveletF1DModule_62912680952299
// MI455X (gfx1250) — compile-verified
//
#include <hip/hip_runtime.h>
#include <math.h>

// Problem constants (fixed by setup_inputs): B=32, N=512, M=512, 2 levels, 32 modes.
#define BATCH   32
#define MCH     512
#define NFULL   512
#define RTOT    (BATCH * MCH)     // 16384 rows, channel-major: row = m*32 + b
#define S257    272               // padded stride for length-257 signals (mult of 16)
#define S130    144               // padded stride for length-130 signals (mult of 16)

// sym2 filters
__constant__ float c_dec_lo[4] = {-0.12940952255092145f, 0.22414386804185735f, 0.836516303737469f, 0.48296291314469025f};
__constant__ float c_dec_hi[4] = {-0.48296291314469025f, 0.836516303737469f, -0.22414386804185735f, -0.12940952255092145f};
__constant__ float c_rec_lo[4] = { 0.48296291314469025f, 0.836516303737469f, 0.22414386804185735f, -0.12940952255092145f};
__constant__ float c_rec_hi[4] = {-0.12940952255092145f, -0.22414386804185735f, 0.836516303737469f, -0.48296291314469025f};

typedef __attribute__((ext_vector_type(2))) float v2f;
typedef __attribute__((ext_vector_type(8))) float v8f;

__device__ inline v8f wmma_f32_4(v2f a, v2f b, v8f c) {
  // V_WMMA_F32_16X16X4_F32 : D(16x16 f32) = A(16x4 f32) * B(4x16 f32) + C
  return __builtin_amdgcn_wmma_f32_16x16x4_f32(false, a, false, b, (short)0, c, false, false);
}

// ---------------------------------------------------------------------------
// Packed DFT basis, K-pair interleaved for single b64 B-fragment loads:
//   bfp[((t>>1)*64 + c)*2 + (t&1)] = (t<N) ? (c<32 ? cos(2pi*k*t/N) : -sin) : 0
//   (k = c&31), allocated for t in [0, Spad)
// ---------------------------------------------------------------------------
__global__ void init_bfp_kernel(float* __restrict__ bfp, int N, int Spad) {
  int idx = blockIdx.x * blockDim.x + threadIdx.x;
  if (idx >= Spad * 64) return;
  // idx = ((t>>1)*64 + c)*2 + (t&1)
  int tl = idx & 1;
  int c  = (idx >> 1) & 63;
  int tp = idx >> 7;
  int t  = tp * 2 + tl;
  int k  = c & 31;
  float v = 0.f;
  if (t < N) {
    int r = (k * t) % N;   // exact periodic reduction
    float th = 6.28318530717958647692f * (float)r / (float)N;
    v = (c < 32) ? cosf(th) : -sinf(th);
  }
  bfp[idx] = v;
}

// ---------------------------------------------------------------------------
// Packed inverse basis: gip[((c>>1)*Spad + t)*2 + (c&1)] = Gi[c][t]
//   Gi[c<32][t] = (k?2:1)*cos(2pi*k*t/N)/N ; Gi[32+k][t] = (k? -2*sin/N : 0)
//   zero for t >= N
// ---------------------------------------------------------------------------
__global__ void init_gip_kernel(float* __restrict__ gip, int N, int Spad) {
  int idx = blockIdx.x * blockDim.x + threadIdx.x;
  if (idx >= 64 * Spad) return;
  // idx = ((c>>1)*Spad + t)*2 + (c&1)
  int cl = idx & 1;
  int t  = (idx >> 1) % Spad;
  int cp = (idx >> 1) / Spad;
  int c  = cp * 2 + cl;
  int k  = c & 31;
  float v = 0.f;
  if (t < N) {
    int r = (k * t) % N;
    float th = 6.28318530717958647692f * (float)r / (float)N;
    float invN = 1.0f / (float)N;
    if (c < 32) v = ((k == 0) ? 1.0f : 2.0f) * cosf(th) * invN;
    else        v = (k == 0) ? 0.0f : (-2.0f * sinf(th) * invN);
  }
  gip[idx] = v;
}

// ---------------------------------------------------------------------------
// Pack per-channel complex weights into real 64x64 block matrix, K-pair packed:
//   Wc = [[Wr, Wi], [-Wi, Wr]] (rows = input modes k, cols = output cc)
//   wcp[((m*32 + (k>>1))*64 + cc)*2 + (k&1)] = Wc[m][k][cc]
// ---------------------------------------------------------------------------
__global__ void pack_wc_kernel(const float* __restrict__ wr, const float* __restrict__ wi,
                               int half, float* __restrict__ wcp) {
  int idx = blockIdx.x * blockDim.x + threadIdx.x;
  if (idx >= MCH * 64 * 64) return;
  // idx = ((m*32 + kp)*64 + cc)*2 + kl
  int kl = idx & 1;
  int cc = (idx >> 1) & 63;
  int kp = (idx >> 7) & 31;
  int m  = idx >> 12;
  int k  = kp * 2 + kl;
  int row = k & 31, col = cc & 31;
  float w = 0.f;
  size_t widx = ((size_t)m * half + row) * half + col;
  if (k < 32) w = (cc < 32) ? wr[widx] : wi[widx];
  else        w = (cc < 32) ? -wi[widx] : wr[widx];
  wcp[idx] = w;
}

// ---------------------------------------------------------------------------
// Transposes: x (B,N,M) <-> channel-major rows (m*32+b, len)
// ---------------------------------------------------------------------------
__global__ void transpose_in_kernel(const float* __restrict__ x, float* __restrict__ xt) {
  int idx = blockIdx.x * blockDim.x + threadIdx.x;
  if (idx >= RTOT * NFULL) return;
  int t = idx & (NFULL - 1);
  int row = idx >> 9;                 // NFULL == 512
  int m = row >> 5, b = row & 31;
  xt[idx] = x[((size_t)b * NFULL + t) * MCH + m];
}

__global__ void transpose_out_kernel(const float* __restrict__ y, float* __restrict__ out) {
  int idx = blockIdx.x * blockDim.x + threadIdx.x;
  if (idx >= BATCH * NFULL * MCH) return;
  int m = idx & (MCH - 1);
  int t = (idx >> 9) & (NFULL - 1);
  int b = idx >> 18;
  out[idx] = y[((size_t)(m * 32 + b)) * NFULL + t];
}

// ---------------------------------------------------------------------------
// DWT: lo/hi[o] = sum_k DEC[3-k] * reflect(x)[2o + k - padL]; pad cols zeroed
// ---------------------------------------------------------------------------
__global__ void dwt_kernel(const float* __restrict__ x, float* __restrict__ lo,
                           float* __restrict__ hi, int N, int inStride,
                           int outsize, int outStride, int padL) {
  int idx = blockIdx.x * blockDim.x + threadIdx.x;
  if (idx >= RTOT * outStride) return;
  int o = idx % outStride;
  if (o >= outsize) { lo[idx] = 0.f; hi[idx] = 0.f; return; }
  int r = idx / outStride;
  const float* xr = x + (size_t)r * inStride;
  float al = 0.f, ah = 0.f;
#pragma unroll
  for (int k = 0; k < 4; ++k) {
    int i = 2 * o + k - padL;
    if (i < 0) i = -i;
    if (i >= N) i = 2 * N - 2 - i;
    float v = xr[i];
    al += c_dec_lo[3 - k] * v;
    ah += c_dec_hi[3 - k] * v;
  }
  lo[idx] = al;
  hi[idx] = ah;
}

// ---------------------------------------------------------------------------
// IDWT: out[t] = sum_k REC[3-k] * zdil_pad[t+k], out len = 2*Nl-2
// ---------------------------------------------------------------------------
__global__ void idwt_kernel(const float* __restrict__ lo, const float* __restrict__ hi,
                            float* __restrict__ out, int Nl, int inStride) {
  int Lout = 2 * Nl - 2;
  int idx = blockIdx.x * blockDim.x + threadIdx.x;
  if (idx >= RTOT * Lout) return;
  int t = idx % Lout, r = idx / Lout;
  const float* lr = lo + (size_t)r * inStride;
  const float* hr = hi + (size_t)r * inStride;
  float acc = 0.f;
#pragma unroll
  for (int k = 0; k < 4; ++k) {
    int i = t + k - 1;
    if (i >= 0 && (i & 1) == 0) {
      int j = i >> 1;
      if (j < Nl) acc += c_rec_lo[3 - k] * lr[j] + c_rec_hi[3 - k] * hr[j];
    }
  }
  out[(size_t)r * Lout + t] = acc;
}

// acc257[r][t] = t258[r][t] + us[r][t]  (crop 258 -> 257 and add; strided out)
__global__ void addcrop_kernel(const float* __restrict__ a258, const float* __restrict__ us,
                               float* __restrict__ out257) {
  int idx = blockIdx.x * blockDim.x + threadIdx.x;
  if (idx >= RTOT * 257) return;
  int t = idx % 257, r = idx / 257;
  out257[(size_t)r * S257 + t] = a258[(size_t)r * 258 + t] + us[(size_t)r * S257 + t];
}

// ---------------------------------------------------------------------------
// WMMA stage 1: Y(RTOT x 64) = sig(RTOT x Npad) * Bf(Npad x 64)  (truncated rDFT)
// Unconditional b64 loads (zero-padded operands), unrolled x4.
// ---------------------------------------------------------------------------
__global__ void dft_kernel(const float* __restrict__ sig, const float* __restrict__ bfp,
                           float* __restrict__ Y, int Npad) {
  int gw = (blockIdx.x * blockDim.x + threadIdx.x) >> 5;
  int lane = threadIdx.x & 31;
  const int nTiles = (RTOT / 16) * 4;
  if (gw >= nTiles) return;
  int tr = gw >> 2, tc = gw & 3;
  int rowBase = tr << 4, colBase = tc << 4;
  int ln = lane & 15;
  int koff = (lane >> 4) << 1;              // 0 or 2
  const float* arow = sig + (size_t)(rowBase + ln) * Npad + koff;
  const float* brow = bfp + (size_t)(koff >> 1) * 128 + (size_t)(colBase + ln) * 2;
  v8f acc = {0.f, 0.f, 0.f, 0.f, 0.f, 0.f, 0.f, 0.f};
  for (int k0 = 0; k0 < Npad; k0 += 16) {
    v2f a0 = *(const v2f*)(arow);
    v2f b0 = *(const v2f*)(brow);
    v2f a1 = *(const v2f*)(arow + 4);
    v2f b1 = *(const v2f*)(brow + 256);
    v2f a2 = *(const v2f*)(arow + 8);
    v2f b2 = *(const v2f*)(brow + 512);
    v2f a3 = *(const v2f*)(arow + 12);
    v2f b3 = *(const v2f*)(brow + 768);
    acc = wmma_f32_4(a0, b0, acc);
    acc = wmma_f32_4(a1, b1, acc);
    acc = wmma_f32_4(a2, b2, acc);
    acc = wmma_f32_4(a3, b3, acc);
    arow += 16;
    brow += 1024;
  }
  int hb = lane >> 4;
#pragma unroll
  for (int v = 0; v < 8; ++v)
    Y[(size_t)(rowBase + v + 8 * hb) * 64 + colBase + ln] = acc[v];
}

// ---------------------------------------------------------------------------
// WMMA stage 2: Z(RTOT x 64) = Y * Wc[m] (64x64 real block form of complex mix)
// Rows are channel-major, so each 16-row tile has a single m = tr>>1.
// ---------------------------------------------------------------------------
__global__ void mix_kernel(const float* __restrict__ Y, const float* __restrict__ wcp,
                           float* __restrict__ Z) {
  int gw = (blockIdx.x * blockDim.x + threadIdx.x) >> 5;
  int lane = threadIdx.x & 31;
  const int nTiles = (RTOT / 16) * 4;
  if (gw >= nTiles) return;
  int tr = gw >> 2, tc = gw & 3;
  int rowBase = tr << 4, colBase = tc << 4;
  int m = tr >> 1;                          // 32 rows (one batch group) per channel
  int ln = lane & 15;
  int koff = (lane >> 4) << 1;
  const float* arow = Y + (size_t)(rowBase + ln) * 64 + koff;
  const float* bp = wcp + ((size_t)(m * 32 + (koff >> 1)) * 64 + colBase + ln) * 2;
  v8f acc = {0.f, 0.f, 0.f, 0.f, 0.f, 0.f, 0.f, 0.f};
#pragma unroll
  for (int k0 = 0; k0 < 64; k0 += 4) {
    v2f a = *(const v2f*)(arow + k0);
    v2f b = *(const v2f*)(bp + (k0 >> 1) * 128);
    acc = wmma_f32_4(a, b, acc);
  }
  int hb = lane >> 4;
#pragma unroll
  for (int v = 0; v < 8; ++v)
    Z[(size_t)(rowBase + v + 8 * hb) * 64 + colBase + ln] = acc[v];
}

// ---------------------------------------------------------------------------
// WMMA stage 3: out(RTOT x Spad) (+)= Z(RTOT x 64) * Gi(64 x Spad) (trunc irDFT)
// Unconditional loads & stores (padded cols of Gi are zero -> pads get zero).
// ---------------------------------------------------------------------------
__global__ void idft_kernel(const float* __restrict__ Z, const float* __restrict__ gip,
                            float* __restrict__ out, int Spad, int nct, int accum) {
  int gw = (blockIdx.x * blockDim.x + threadIdx.x) >> 5;
  int lane = threadIdx.x & 31;
  const int nTiles = (RTOT / 16) * nct;
  if (gw >= nTiles) return;
  int tr = gw / nct, tc = gw % nct;
  int rowBase = tr << 4, colBase = tc << 4;
  int ln = lane & 15;
  int koff = (lane >> 4) << 1;
  int t = colBase + ln;
  const float* arow = Z + (size_t)(rowBase + ln) * 64 + koff;
  const float* bp = gip + ((size_t)(koff >> 1) * Spad + t) * 2;
  v8f acc = {0.f, 0.f, 0.f, 0.f, 0.f, 0.f, 0.f, 0.f};
#pragma unroll
  for (int k0 = 0; k0 < 64; k0 += 4) {
    v2f a = *(const v2f*)(arow + k0);
    v2f b = *(const v2f*)(bp + (size_t)(k0 >> 1) * Spad * 2);
    acc = wmma_f32_4(a, b, acc);
  }
  int hb = lane >> 4;
  if (accum) {
#pragma unroll
    for (int v = 0; v < 8; ++v)
      out[(size_t)(rowBase + v + 8 * hb) * Spad + t] += acc[v];
  } else {
#pragma unroll
    for (int v = 0; v < 8; ++v)
      out[(size_t)(rowBase + v + 8 * hb) * Spad + t] = acc[v];
  }
}

// ---------------------------------------------------------------------------
// Host orchestration
// ---------------------------------------------------------------------------
static inline int ceil_div(int a, int b) { return (a + b - 1) / b; }

static void run_ms(const float* sig, int Spad, const float* wr, const float* wi, int half,
                   float* outbuf, int accum, float* Ybuf, float* Zbuf, float* Wcp,
                   const float* bfp, const float* gip, hipStream_t stream) {
  pack_wc_kernel<<<ceil_div(MCH * 64 * 64, 256), 256, 0, stream>>>(wr, wi, half, Wcp);
  const int tiles64 = (RTOT / 16) * 4;
  dft_kernel<<<ceil_div(tiles64, 4), 128, 0, stream>>>(sig, bfp, Ybuf, Spad);
  mix_kernel<<<ceil_div(tiles64, 4), 128, 0, stream>>>(Ybuf, Wcp, Zbuf);
  int nct = Spad / 16;
  int tilesN = (RTOT / 16) * nct;
  idft_kernel<<<ceil_div(tilesN, 4), 128, 0, stream>>>(Zbuf, gip, outbuf, Spad, nct, accum);
}

extern "C" void kernel_launch(void* const* d_in, const int* in_sizes, int n_in,
                              void* d_out, int out_size, void* d_ws, size_t ws_size,
                              hipStream_t stream) {
  const float* x = (const float*)d_in[0];
  const float* l0k0_r = (const float*)d_in[1];  const float* l0k0_i = (const float*)d_in[2];
  const float* l0k1_r = (const float*)d_in[3];  const float* l0k1_i = (const float*)d_in[4];
  const float* l0k2_r = (const float*)d_in[5];  const float* l0k2_i = (const float*)d_in[6];
  const float* l1k0_r = (const float*)d_in[7];  const float* l1k0_i = (const float*)d_in[8];
  const float* l1k1_r = (const float*)d_in[9];  const float* l1k1_i = (const float*)d_in[10];
  const float* l1k2_r = (const float*)d_in[11]; const float* l1k2_i = (const float*)d_in[12];
  const float* D_r    = (const float*)d_in[13]; const float* D_i    = (const float*)d_in[14];
  float* out = (float*)d_out;

  int half0 = (int)llround(sqrt((double)in_sizes[1]  / (double)MCH));   // 86
  int half1 = (int)llround(sqrt((double)in_sizes[7]  / (double)MCH));   // 65
  int halfD = (int)llround(sqrt((double)in_sizes[13] / (double)MCH));   // 132

  // Workspace carve-up (256B aligned)
  size_t off = 0;
  char* base = (char*)d_ws;
  auto alloc = [&](size_t nfloats) -> float* {
    float* p = (float*)(base + off);
    off += ((nfloats * sizeof(float)) + 255) & ~(size_t)255;
    return p;
  };
  float* xt     = alloc((size_t)RTOT * NFULL);   // reused as y512 at the end
  float* lo0    = alloc((size_t)RTOT * S257);
  float* hi0    = alloc((size_t)RTOT * S257);
  float* Ud0    = alloc((size_t)RTOT * S257);
  float* Us1    = alloc((size_t)RTOT * S257);
  float* lo1    = alloc((size_t)RTOT * S130);
  float* hi1    = alloc((size_t)RTOT * S130);
  float* Ud1    = alloc((size_t)RTOT * S130);
  float* acc130 = alloc((size_t)RTOT * S130);
  float* t258   = alloc((size_t)RTOT * 258);
  float* acc257 = alloc((size_t)RTOT * S257);
  float* Ybuf   = alloc((size_t)RTOT * 64);
  float* Zbuf   = alloc((size_t)RTOT * 64);
  float* Wcp    = alloc((size_t)MCH * 64 * 64);
  float* bfp257 = alloc((size_t)S257 * 64);
  float* gip257 = alloc((size_t)64 * S257);
  float* bfp130 = alloc((size_t)S130 * 64);
  float* gip130 = alloc((size_t)64 * S130);
  (void)ws_size; (void)n_in; (void)out_size;

  // Packed bases (zero-padded beyond valid length)
  init_bfp_kernel<<<ceil_div(S257 * 64, 256), 256, 0, stream>>>(bfp257, 257, S257);
  init_gip_kernel<<<ceil_div(64 * S257, 256), 256, 0, stream>>>(gip257, 257, S257);
  init_bfp_kernel<<<ceil_div(S130 * 64, 256), 256, 0, stream>>>(bfp130, 130, S130);
  init_gip_kernel<<<ceil_div(64 * S130, 256), 256, 0, stream>>>(gip130, 130, S130);

  // (B,N,M) -> channel-major rows (m*32+b, 512)
  transpose_in_kernel<<<ceil_div(RTOT * NFULL, 256), 256, 0, stream>>>(x, xt);

  // Level 0 DWT: 512 -> 257 (pad p=4, padL=2), out stride 272, pads zeroed
  dwt_kernel<<<ceil_div(RTOT * S257, 256), 256, 0, stream>>>(xt, lo0, hi0, 512, 512, 257, S257, 2);

  // Level 0 spectral kernels (N=257, Spad=272)
  run_ms(hi0, S257, l0k0_r, l0k0_i, half0, Ud0, 0, Ybuf, Zbuf, Wcp, bfp257, gip257, stream);
  run_ms(lo0, S257, l0k1_r, l0k1_i, half0, Ud0, 1, Ybuf, Zbuf, Wcp, bfp257, gip257, stream);
  run_ms(lo0, S257, l0k2_r, l0k2_i, half0, Us1, 0, Ybuf, Zbuf, Wcp, bfp257, gip257, stream);

  // Level 1 DWT: 257 -> 130 (pad p=5, padL=2), out stride 144, pads zeroed
  dwt_kernel<<<ceil_div(RTOT * S130, 256), 256, 0, stream>>>(lo0, lo1, hi1, 257, S257, 130, S130, 2);

  // Level 1 spectral kernels (N=130, Spad=144)
  run_ms(hi1, S130, l1k0_r, l1k0_i, half1, Ud1, 0, Ybuf, Zbuf, Wcp, bfp130, gip130, stream);
  run_ms(lo1, S130, l1k1_r, l1k1_i, half1, Ud1, 1, Ybuf, Zbuf, Wcp, bfp130, gip130, stream);
  run_ms(lo1, S130, D_r,    D_i,    halfD, acc130, 0, Ybuf, Zbuf, Wcp, bfp130, gip130, stream); // x = ms(lo1, D)
  run_ms(lo1, S130, l1k2_r, l1k2_i, half1, acc130, 1, Ybuf, Zbuf, Wcp, bfp130, gip130, stream); // x += Us[2]

  // Reconstruction
  idwt_kernel<<<ceil_div(RTOT * 258, 256), 256, 0, stream>>>(acc130, Ud1, t258, 130, S130);   // -> len 258
  addcrop_kernel<<<ceil_div(RTOT * 257, 256), 256, 0, stream>>>(t258, Us1, acc257);           // crop+add (stride S257)
  idwt_kernel<<<ceil_div(RTOT * 512, 256), 256, 0, stream>>>(acc257, Ud0, xt, 257, S257);     // -> len 512 (reuse xt)

  // channel-major -> (B,N,M)
  transpose_out_kernel<<<ceil_div(BATCH * NFULL * MCH, 256), 256, 0, stream>>>(xt, out);
}